// TensorRingGaussian_54966991454794
// MI455X (gfx1250) — compile-verified
//
#include <hip/hip_runtime.h>
#include <math.h>

typedef float v2f __attribute__((ext_vector_type(2)));
typedef float v8f __attribute__((ext_vector_type(8)));

#define KK 16
#define MM 64
#define WPB 8               // waves per block
#define BLOCK (WPB * 32)

// lane <-> lane^16 half-wave swap via ds_swizzle (SWAPX16: xor=0x10, and=0x1f)
__device__ __forceinline__ float swap16(float x) {
    return __int_as_float(__builtin_amdgcn_ds_swizzle(__float_as_int(x), 0x401f));
}

__device__ __forceinline__ float softplus_f(float p) {
    return log1pf(__expf(-fabsf(p))) + fmaxf(p, 0.0f);
}

__launch_bounds__(BLOCK)
__global__ void trg_kernel(const float* __restrict__ X,
                           const float* __restrict__ W,
                           const float* __restrict__ mu,
                           const float* __restrict__ psig,
                           float* __restrict__ out, int Btot)
{
    // [m][lane][e]: logW(+log_norm) reordered into WMMA A-operand order
    __shared__ __attribute__((aligned(16))) float ls_c1[MM * 32 * 8];   // 64 KB
    __shared__ float ls_lse[MM * KK];                                   // 4 KB
    __shared__ float ls_logn[KK * KK];
    __shared__ float ls_mu[KK * KK];
    __shared__ float ls_is[KK * KK];
    __shared__ __attribute__((aligned(16))) float ls_c0t[32 * 8];       // dim-0 transposed tables
    __shared__ __attribute__((aligned(16))) float ls_mut[32 * 8];
    __shared__ __attribute__((aligned(16))) float ls_ist[32 * 8];
    __shared__ float ls_x[WPB * MM];

    const int tid = threadIdx.x;

    // ---- per-(i,j) parameter prep: sigma, 1/sigma, log-normalizer (256 threads, 1 each)
    {
        float sig = softplus_f(psig[tid]);
        ls_mu[tid]   = mu[tid];
        ls_is[tid]   = 1.0f / sig;
        ls_logn[tid] = -__logf(sig) - 0.91893853320467274f;   // -log(sigma) - 0.5*log(2*pi)
    }
    // ---- logsumexp over axis=1 (row index i) for each (m, j)
    for (int n = tid; n < MM * KK; n += BLOCK) {
        int m = n >> 4, j = n & 15;
        const float* w = W + m * (KK * KK) + j;   // stride KK over i
        float mx = w[0];
        #pragma unroll
        for (int i = 1; i < KK; ++i) mx = fmaxf(mx, w[i * KK]);
        float s = 0.0f;
        #pragma unroll
        for (int i = 0; i < KK; ++i) s += __expf(w[i * KK] - mx);
        ls_lse[n] = mx + __logf(s);
    }
    __syncthreads();
    // ---- C1[m][lane][e] = logW[m,i,j] + log_norm[i,j] in A-operand order
    //      lane = h*16+i (h = K-half), e = 2c+o -> j = 4c + 2h + o
    for (int n = tid; n < MM * 256; n += BLOCK) {
        int m = n >> 8, rem = n & 255;
        int lane = rem >> 3, e = rem & 7;
        int i = lane & 15, h = lane >> 4;
        int j = 4 * (e >> 1) + 2 * h + (e & 1);
        ls_c1[n] = W[m * (KK * KK) + i * KK + j] - ls_lse[m * KK + j] + ls_logn[i * KK + j];
    }
    // ---- transposed tables for the final (dim 0, un-transposed) factor: A = R0^T
    {
        int lane = tid >> 3, e = tid & 7;
        int h = lane >> 4;
        int iT = 4 * (e >> 1) + 2 * h + (e & 1);
        int jT = lane & 15;
        int idx = iT * KK + jT;
        ls_c0t[tid] = W[idx] - ls_lse[jT] + ls_logn[idx];
        ls_mut[tid] = ls_mu[idx];
        ls_ist[tid] = ls_is[idx];
    }

    // ---- stage this wave's X row into LDS (broadcast source for per-step scalar x)
    const int wave = tid >> 5, lane = tid & 31;
    long long b = (long long)blockIdx.x * WPB + wave;
    if (b >= Btot) b = Btot - 1;              // wave-uniform clamp; duplicate writes are identical
    const float* xr = X + b * MM;
    ls_x[wave * MM + lane]      = xr[lane];
    ls_x[wave * MM + 32 + lane] = xr[32 + lane];
    __syncthreads();

    // per-lane gaussian constants for the A-operand entry positions (fixed across steps)
    const int i0 = lane & 15, h = lane >> 4;
    float muv[8], isv[8];
    #pragma unroll
    for (int e = 0; e < 8; ++e) {
        int j = 4 * (e >> 1) + 2 * h + (e & 1);
        muv[e] = ls_mu[i0 * KK + j];
        isv[e] = ls_is[i0 * KK + j];
    }

    // Q = identity in WMMA C/D layout: lane<16: Q[v,lane]; lane>=16: Q[8+v,lane-16]
    v8f acc;
    #pragma unroll
    for (int v = 0; v < 8; ++v)
        acc[v] = (lane < 16) ? ((v == lane) ? 1.0f : 0.0f)
                             : (((v + 8) == (lane - 16)) ? 1.0f : 0.0f);

    const bool lo = (lane < 16);
    const float* xrow = ls_x + wave * MM;

    for (int m = 1; m < MM; ++m) {
        float xs = xrow[m];                                   // broadcast DS read
        const float4* cp = (const float4*)(ls_c1 + m * 256 + lane * 8);
        float4 ca = cp[0], cb = cp[1];
        float c1v[8] = {ca.x, ca.y, ca.z, ca.w, cb.x, cb.y, cb.z, cb.w};
        float r[8];
        #pragma unroll
        for (int e = 0; e < 8; ++e) {
            float z = (xs - muv[e]) * isv[e];
            r[e] = __expf(__builtin_fmaf(-0.5f * z, z, c1v[e]));
        }
        // A chunks: R[:,4c:4c+4]
        v2f A0 = {r[0], r[1]}, A1 = {r[2], r[3]}, A2 = {r[4], r[5]}, A3 = {r[6], r[7]};
        // B chunks: Q[4c:4c+4,:] rebuilt from D layout via half-wave swaps
        float s0 = swap16(acc[0]), s1 = swap16(acc[1]), s2 = swap16(acc[2]), s3 = swap16(acc[3]);
        float s4 = swap16(acc[4]), s5 = swap16(acc[5]), s6 = swap16(acc[6]), s7 = swap16(acc[7]);
        v2f B0, B1, B2, B3;
        B0[0] = lo ? acc[0] : s2;  B0[1] = lo ? acc[1] : s3;   // rows 0,2 / 1,3
        B1[0] = lo ? acc[4] : s6;  B1[1] = lo ? acc[5] : s7;   // rows 4,6 / 5,7
        B2[0] = lo ? s0 : acc[2];  B2[1] = lo ? s1 : acc[3];   // rows 8,10 / 9,11
        B3[0] = lo ? s4 : acc[6];  B3[1] = lo ? s5 : acc[7];   // rows 12,14 / 13,15
        v8f d = {0.f,0.f,0.f,0.f,0.f,0.f,0.f,0.f};
        d = __builtin_amdgcn_wmma_f32_16x16x4_f32(false, A0, false, B0, (short)0, d, false, false);
        d = __builtin_amdgcn_wmma_f32_16x16x4_f32(false, A1, false, B1, (short)0, d, false, false);
        d = __builtin_amdgcn_wmma_f32_16x16x4_f32(false, A2, false, B2, (short)0, d, false, false);
        d = __builtin_amdgcn_wmma_f32_16x16x4_f32(false, A3, false, B3, (short)0, d, false, false);
        acc = d;                                              // Q <- R_m * Q
    }

    // final factor: sum(P*R0) = sum(R0^T * Q); one more step with A = R0^T
    {
        float xs = xrow[0];
        const float4* cp = (const float4*)(ls_c0t + lane * 8);
        const float4* mp = (const float4*)(ls_mut + lane * 8);
        const float4* ip = (const float4*)(ls_ist + lane * 8);
        float4 ca = cp[0], cb = cp[1];
        float4 ma = mp[0], mb = mp[1];
        float4 ia = ip[0], ib = ip[1];
        float c1v[8] = {ca.x, ca.y, ca.z, ca.w, cb.x, cb.y, cb.z, cb.w};
        float mt[8]  = {ma.x, ma.y, ma.z, ma.w, mb.x, mb.y, mb.z, mb.w};
        float it[8]  = {ia.x, ia.y, ia.z, ia.w, ib.x, ib.y, ib.z, ib.w};
        float r[8];
        #pragma unroll
        for (int e = 0; e < 8; ++e) {
            float z = (xs - mt[e]) * it[e];
            r[e] = __expf(__builtin_fmaf(-0.5f * z, z, c1v[e]));
        }
        v2f A0 = {r[0], r[1]}, A1 = {r[2], r[3]}, A2 = {r[4], r[5]}, A3 = {r[6], r[7]};
        float s0 = swap16(acc[0]), s1 = swap16(acc[1]), s2 = swap16(acc[2]), s3 = swap16(acc[3]);
        float s4 = swap16(acc[4]), s5 = swap16(acc[5]), s6 = swap16(acc[6]), s7 = swap16(acc[7]);
        v2f B0, B1, B2, B3;
        B0[0] = lo ? acc[0] : s2;  B0[1] = lo ? acc[1] : s3;
        B1[0] = lo ? acc[4] : s6;  B1[1] = lo ? acc[5] : s7;
        B2[0] = lo ? s0 : acc[2];  B2[1] = lo ? s1 : acc[3];
        B3[0] = lo ? s4 : acc[6];  B3[1] = lo ? s5 : acc[7];
        v8f d = {0.f,0.f,0.f,0.f,0.f,0.f,0.f,0.f};
        d = __builtin_amdgcn_wmma_f32_16x16x4_f32(false, A0, false, B0, (short)0, d, false, false);
        d = __builtin_amdgcn_wmma_f32_16x16x4_f32(false, A1, false, B1, (short)0, d, false, false);
        d = __builtin_amdgcn_wmma_f32_16x16x4_f32(false, A2, false, B2, (short)0, d, false, false);
        d = __builtin_amdgcn_wmma_f32_16x16x4_f32(false, A3, false, B3, (short)0, d, false, false);
        acc = d;
    }

    // total = sum of all 256 entries, then log(total + eps)
    float tot = acc[0] + acc[1] + acc[2] + acc[3] + acc[4] + acc[5] + acc[6] + acc[7];
    #pragma unroll
    for (int off = 16; off > 0; off >>= 1)
        tot += __shfl_xor(tot, off, 32);
    if (lane == 0)
        out[b] = __logf(tot + 2.220446049250313e-16f);
}

extern "C" void kernel_launch(void* const* d_in, const int* in_sizes, int n_in,
                              void* d_out, int out_size, void* d_ws, size_t ws_size,
                              hipStream_t stream) {
    const float* X  = (const float*)d_in[0];   // (B, 64)
    const float* W  = (const float*)d_in[1];   // (64, 16, 16)
    const float* mu = (const float*)d_in[2];   // (16, 16)
    const float* ps = (const float*)d_in[3];   // (16, 16)
    float* out = (float*)d_out;                // (B,)
    const int Btot = in_sizes[0] / MM;
    const int blocks = (Btot + WPB - 1) / WPB;
    trg_kernel<<<blocks, BLOCK, 0, stream>>>(X, W, mu, ps, out, Btot);
}